// cluster_similarity_loss_446676599061
// MI455X (gfx1250) — compile-verified
//
#include <hip/hip_runtime.h>
#include <hip/hip_bf16.h>

typedef __attribute__((ext_vector_type(2))) float v2f;
typedef __attribute__((ext_vector_type(8))) float v8f;

#define BS 16
#define NV 2048
#define NC 16
#define NTOT 67108864ull            // 16*2048*2048 = 2^26
#define MAIN_BLOCKS 2048
#define THREADS 256
#define WAVES_PER_BLOCK (THREADS / 32)
#define ELEMS_PER_WAVE 4096         // NTOT / (MAIN_BLOCKS * WAVES_PER_BLOCK)
#define MAIN_ITERS 32               // ELEMS_PER_WAVE / (32 lanes * 4 floats)
#define N_PARTIALS (MAIN_BLOCKS + BS)

// ---------------------------------------------------------------------------
// Kernel 1: streaming sum of squares over all 2^26 floats.
// Each wave owns a contiguous 4096-element span; float4 (b128) loads saturate
// HBM. Per-lane partials are reduced across the wave with a single
// V_WMMA_F32_16X16X4_F32 against an all-ones B matrix.
// ---------------------------------------------------------------------------
__global__ __launch_bounds__(THREADS)
void sumsq_kernel(const float* __restrict__ x, double* __restrict__ partials) {
    const int lane = threadIdx.x & 31;
    const int wave = threadIdx.x >> 5;
    const unsigned gwave = blockIdx.x * WAVES_PER_BLOCK + wave;
    const size_t base = (size_t)gwave * ELEMS_PER_WAVE;

    const float4* __restrict__ src = (const float4*)(x + base) + lane;

    float p0 = 0.0f, p1 = 0.0f;
#pragma unroll 4
    for (int it = 0; it < MAIN_ITERS; ++it) {
        float4 v = src[it * 32];
        // gfx1250 global_prefetch_b8: stay ~4KB ahead of the load stream.
        __builtin_prefetch((const void*)(src + (it + 8) * 32), 0, 0);
        p0 = fmaf(v.x, v.x, p0);
        p0 = fmaf(v.y, v.y, p0);
        p1 = fmaf(v.z, v.z, p1);
        p1 = fmaf(v.w, v.w, p1);
    }

    // WMMA wave reduction. A = 16x4 f32 tile holding all 64 lane partials
    // (2 VGPRs/lane across 32 lanes), B = 4x16 ones. D[m][n] = rowsum_m.
    // Uniform control flow above => EXEC is all ones here (WMMA requirement).
    v2f a = {p0, p1};
    v2f ones = {1.0f, 1.0f};
    v8f c = {};
    c = __builtin_amdgcn_wmma_f32_16x16x4_f32(
        /*neg_a=*/false, a, /*neg_b=*/false, ones,
        /*c_mod=*/(short)0, c, /*reuse_a=*/false, /*reuse_b=*/false);

    // Lanes 0-15 hold rows 0-7 in their 8 C VGPRs; lanes 16-31 hold rows 8-15.
    float s = c[0] + c[1] + c[2] + c[3] + c[4] + c[5] + c[6] + c[7];
    s += __shfl_xor(s, 16, 32);   // combine the two half-wave row groups

    __shared__ float wsum[WAVES_PER_BLOCK];
    if (lane == 0) wsum[wave] = s;
    __syncthreads();
    if (threadIdx.x == 0) {
        float bs = 0.0f;
#pragma unroll
        for (int i = 0; i < WAVES_PER_BLOCK; ++i) bs += wsum[i];
        partials[blockIdx.x] = (double)bs;
    }
}

// ---------------------------------------------------------------------------
// Kernel 2: correction term  sum_{target==1} (1 - 2*x)  over the 256x256
// corner of each batch. Block b handles batch b. Interval membership of index
// i in cluster k (start_k = n_{k-1}, start_0 = 0) is encoded as a 16-bit mask;
// target(i,j) = (mask[i] & mask[j]) != 0, which collapses overlaps to 1
// exactly like the reference's (cov > 0).
// ---------------------------------------------------------------------------
__global__ __launch_bounds__(256)
void corr_kernel(const float* __restrict__ x, const int* __restrict__ cs,
                 double* __restrict__ partials) {
    const int b = blockIdx.x;
    const int t = threadIdx.x;   // index 0..255 (covers all possible blocks)

    __shared__ unsigned mask[256];
    unsigned m = 0;
    int prev = 0;
#pragma unroll
    for (int k = 0; k < NC; ++k) {
        int n = cs[b * NC + k];
        int s = (k == 0) ? 0 : prev;   // start_k = n_{k-1}  (NOT cumulative)
        int e = s + n;
        if (t >= s && t < e) m |= (1u << k);
        prev = n;
    }
    mask[t] = m;
    __syncthreads();

    float corr = 0.0f;
    const unsigned mi = mask[t];
    if (mi) {
        const float* row = x + ((size_t)b * NV + t) * NV;
        for (int j = 0; j < 256; ++j) {
            if (mi & mask[j]) corr += 1.0f - 2.0f * row[j];
        }
    }

    __shared__ float red[256];
    red[t] = corr;
    __syncthreads();
    for (int off = 128; off > 0; off >>= 1) {
        if (t < off) red[t] += red[t + off];
        __syncthreads();
    }
    if (t == 0) partials[MAIN_BLOCKS + b] = (double)red[0];
}

// ---------------------------------------------------------------------------
// Kernel 3: deterministic final reduction of 2064 double partials.
// ---------------------------------------------------------------------------
__global__ __launch_bounds__(256)
void finalize_kernel(const double* __restrict__ partials, float* __restrict__ out) {
    __shared__ double red[256];
    double s = 0.0;
    for (int i = threadIdx.x; i < N_PARTIALS; i += 256) s += partials[i];
    red[threadIdx.x] = s;
    __syncthreads();
    for (int off = 128; off > 0; off >>= 1) {
        if (threadIdx.x < off) red[threadIdx.x] += red[threadIdx.x + off];
        __syncthreads();
    }
    if (threadIdx.x == 0) out[0] = (float)(red[0] / (double)NTOT);
}

extern "C" void kernel_launch(void* const* d_in, const int* in_sizes, int n_in,
                              void* d_out, int out_size, void* d_ws, size_t ws_size,
                              hipStream_t stream) {
    const float* raw_scores    = (const float*)d_in[0];
    const int*   cluster_sizes = (const int*)d_in[1];
    double* partials = (double*)d_ws;   // N_PARTIALS doubles, all written each call

    sumsq_kernel<<<MAIN_BLOCKS, THREADS, 0, stream>>>(raw_scores, partials);
    corr_kernel<<<BS, 256, 0, stream>>>(raw_scores, cluster_sizes, partials);
    finalize_kernel<<<1, 256, 0, stream>>>(partials, (float*)d_out);
}